// GENIE_network_88794153878166
// MI455X (gfx1250) — compile-verified
//
#include <hip/hip_runtime.h>

// GENIE network forward for LAT=(32,16,16,16), npaths=35, fp32 throughout.
// Dense 35x35 channel-mix layers use V_WMMA_F32_16X16X4_F32 (fp32 matrix pipe,
// no precision loss, no convert traffic); each wave owns 4 column tiles so the
// LDS weight fragments feed 4 WMMA chains each. Transports / link doubling are
// streaming 3x3 matmuls (memory bound -> VALU fp32, b128 vectorized).
//
// Workspace layout (floats):
//   bufA : 35 * NC            (220.2 MB)  activation ping
//   bufB : 35 * NC            (220.2 MB)  activation pong
//   mats : 34 * NSITE * 9     (160.4 MB)  transport matrices, path-indexed
//   wpad : 3 * 48 * 36        (padded W1..W3 for WMMA tiles)

#define NSITE 131072L            // 32*16*16*16
#define NC    1572864L           // NSITE * 12 (4 spin x 3 color)
#define NPATH 35

typedef float v2f __attribute__((ext_vector_type(2)));
typedef float v8f __attribute__((ext_vector_type(8)));

struct PathTab { short mu[NPATH]; short n[NPATH]; };

__device__ __forceinline__ int shift_site(int site, int mu, int d) {
  // flat site = ((t*16 + x)*16 + y)*16 + z ; periodic wrap, dims are pow2
  int z = site & 15, y = (site >> 4) & 15, x = (site >> 8) & 15, t = site >> 12;
  if (mu == 0)      t = (t + d + 32) & 31;
  else if (mu == 1) x = (x + d + 16) & 15;
  else if (mu == 2) y = (y + d + 16) & 15;
  else              z = (z + d + 16) & 15;
  return (t << 12) | (x << 8) | (y << 4) | z;
}

// ---- pad W1..W3 (35x35) into 48x36 zero-padded tiles -----------------------
__global__ __launch_bounds__(256) void genie_padw(
    const float* __restrict__ W1, const float* __restrict__ W2,
    const float* __restrict__ W3, float* __restrict__ wpad) {
  const int tid = blockIdx.x * 256 + threadIdx.x;
  if (tid >= 3 * 48 * 36) return;
  const int l = tid / 1728, rem = tid % 1728;
  const int o = rem / 36, i = rem % 36;
  const float* W = (l == 0) ? W1 : ((l == 1) ? W2 : W3);
  wpad[tid] = (o < NPATH && i < NPATH) ? W[o * NPATH + i] : 0.0f;
}

// ---- transport matrices, length 1: forward = U_mu(x), backward = U_mu(x-mu)^T
__global__ __launch_bounds__(256) void genie_matinit(
    const float* __restrict__ U, float* __restrict__ mf,
    float* __restrict__ mb, int mu) {
  const int site = blockIdx.x * 256 + threadIdx.x;
  const float* uf = U + ((long)mu * NSITE + site) * 9;
  float* f = mf + (long)site * 9;
#pragma unroll
  for (int i = 0; i < 9; ++i) f[i] = uf[i];
  const int site2 = shift_site(site, mu, -1);
  const float* ub = U + ((long)mu * NSITE + site2) * 9;
  float* b = mb + (long)site * 9;
#pragma unroll
  for (int a = 0; a < 3; ++a)
#pragma unroll
    for (int c = 0; c < 3; ++c)
      b[a * 3 + c] = ub[c * 3 + a];
}

// ---- doubling: M_dst(x) = M_src(x) @ M_src(x + sh*mu_hat) ------------------
__global__ __launch_bounds__(256) void genie_matdouble(
    const float* __restrict__ srcm, float* __restrict__ dstm, int mu, int sh) {
  const int site = blockIdx.x * 256 + threadIdx.x;
  const int site2 = shift_site(site, mu, sh);
  const float* A = srcm + (long)site * 9;
  const float* B = srcm + (long)site2 * 9;
  float c[9];
#pragma unroll
  for (int i = 0; i < 3; ++i)
#pragma unroll
    for (int j = 0; j < 3; ++j)
      c[i * 3 + j] = A[i * 3 + 0] * B[0 * 3 + j] +
                     A[i * 3 + 1] * B[1 * 3 + j] +
                     A[i * 3 + 2] * B[2 * 3 + j];
  float* D = dstm + (long)site * 9;
#pragma unroll
  for (int i = 0; i < 9; ++i) D[i] = c[i];
}

// ---- layer 0: out[o] = W0[o]*v (+ v on o==0), 4 columns/thread (b128) ------
__global__ __launch_bounds__(256) void genie_layer0(
    const float* __restrict__ v, const float* __restrict__ W0,
    float* __restrict__ dst) {
  const long c = ((long)blockIdx.x * 256 + threadIdx.x) * 4;
  const float4 val = *(const float4*)(v + c);
  float4 o0;
  o0.x = W0[0] * val.x + val.x; o0.y = W0[0] * val.y + val.y;
  o0.z = W0[0] * val.z + val.z; o0.w = W0[0] * val.w + val.w;
  *(float4*)(dst + c) = o0;
#pragma unroll
  for (int o = 1; o < NPATH; ++o) {
    const float w = W0[o];
    float4 r; r.x = w * val.x; r.y = w * val.y; r.z = w * val.z; r.w = w * val.w;
    *(float4*)(dst + (long)o * NC + c) = r;
  }
}

// ---- parallel transport: dst[p](x) = M_p(x) @ src[p](x + n_p*mu_p) ---------
__global__ __launch_bounds__(256) void genie_transport(
    const float* __restrict__ src, float* __restrict__ dst,
    const float* __restrict__ mats, PathTab pt) {
  const int p = blockIdx.y;
  const int site = blockIdx.x * 256 + threadIdx.x;
  if (p == 0) {  // empty path: identity copy (48B per site, b128 x3)
    const float4* s = (const float4*)(src + (long)site * 12);
    float4* d = (float4*)(dst + (long)site * 12);
    d[0] = s[0]; d[1] = s[1]; d[2] = s[2];
    return;
  }
  const int mu = pt.mu[p];
  const int sh = pt.n[p];
  const int site2 = shift_site(site, mu, sh);
  const float* __restrict__ M = mats + ((long)(p - 1) * NSITE + site) * 9;
  const float4* __restrict__ vi =
      (const float4*)(src + ((long)p * NSITE + site2) * 12);
  float4* __restrict__ vo = (float4*)(dst + ((long)p * NSITE + site) * 12);
  const float m00 = M[0], m01 = M[1], m02 = M[2];
  const float m10 = M[3], m11 = M[4], m12 = M[5];
  const float m20 = M[6], m21 = M[7], m22 = M[8];
  const float4 i0 = vi[0], i1 = vi[1], i2 = vi[2];
  // 12 contiguous floats = 4 spins x 3 colors; out[s,a] = sum_b M[a,b] v[s,b]
  const float b00 = i0.x, b01 = i0.y, b02 = i0.z;   // s=0
  const float b10 = i0.w, b11 = i1.x, b12 = i1.y;   // s=1
  const float b20 = i1.z, b21 = i1.w, b22 = i2.x;   // s=2
  const float b30 = i2.y, b31 = i2.z, b32 = i2.w;   // s=3
  float4 o0, o1, o2;
  o0.x = m00 * b00 + m01 * b01 + m02 * b02;
  o0.y = m10 * b00 + m11 * b01 + m12 * b02;
  o0.z = m20 * b00 + m21 * b01 + m22 * b02;
  o0.w = m00 * b10 + m01 * b11 + m02 * b12;
  o1.x = m10 * b10 + m11 * b11 + m12 * b12;
  o1.y = m20 * b10 + m21 * b11 + m22 * b12;
  o1.z = m00 * b20 + m01 * b21 + m02 * b22;
  o1.w = m10 * b20 + m11 * b21 + m12 * b22;
  o2.x = m20 * b20 + m21 * b21 + m22 * b22;
  o2.y = m00 * b30 + m01 * b31 + m02 * b32;
  o2.z = m10 * b30 + m11 * b31 + m12 * b32;
  o2.w = m20 * b30 + m21 * b31 + m22 * b32;
  vo[0] = o0; vo[1] = o1; vo[2] = o2;
}

// ---- dense channel mix via V_WMMA_F32_16X16X4_F32 --------------------------
// One wave owns FOUR 16-column tiles (64 columns): the LDS A fragments feed 4
// independent WMMA chains (12 accumulators), amortizing ds traffic and hiding
// matrix-pipe latency. M padded 35->48 (3 tiles), K padded 35->36 (9 steps).
// Layouts follow ISA 7.12.2: lanes 0-15 / 16-31 split K (A,B) and M+8 (C/D).
__global__ __launch_bounds__(256) void genie_dense_wmma(
    const float* __restrict__ src, const float* __restrict__ wpad,
    float* __restrict__ dst) {
  __shared__ float Ws[48 * 36];
  for (int i = threadIdx.x; i < 48 * 36; i += 256) Ws[i] = wpad[i];
  __syncthreads();

  const int lane = threadIdx.x & 31;
  const int wave = (int)((blockIdx.x * 256u + threadIdx.x) >> 5);
  const int m    = lane & 15;        // A row in tile / B-D column in tile
  const int kh   = lane >> 4;        // VGPR-half -> K offset (A,B), M+8 (C/D)
  const long col = (long)wave * 64 + m;   // tile j adds j*16

  v8f acc[3][4] = {};
#pragma unroll
  for (int k = 0; k < 9; ++k) {
    const int k0 = 4 * k + 2 * kh;   // input channel for .x ; .y is k0+1
    v2f b[4];
#pragma unroll
    for (int j = 0; j < 4; ++j) {
      b[j].x = (k0     < NPATH) ? src[(long)k0       * NC + col + j * 16] : 0.0f;
      b[j].y = (k0 + 1 < NPATH) ? src[(long)(k0 + 1) * NC + col + j * 16] : 0.0f;
    }
#pragma unroll
    for (int t = 0; t < 3; ++t) {
      v2f a;
      a.x = Ws[(16 * t + m) * 36 + k0];
      a.y = Ws[(16 * t + m) * 36 + k0 + 1];
#pragma unroll
      for (int j = 0; j < 4; ++j)
        acc[t][j] = __builtin_amdgcn_wmma_f32_16x16x4_f32(
            false, a, false, b[j], (short)0, acc[t][j], false, false);
    }
  }

  float res[4];
#pragma unroll
  for (int j = 0; j < 4; ++j) res[j] = src[col + j * 16];  // ch 0 residual
#pragma unroll
  for (int t = 0; t < 3; ++t) {
#pragma unroll
    for (int r = 0; r < 8; ++r) {
      const int row = 16 * t + r + 8 * kh;
      if (row < NPATH) {
#pragma unroll
        for (int j = 0; j < 4; ++j) {
          float vv = acc[t][j][r];
          if (row == 0) vv += res[j];
          dst[(long)row * NC + col + j * 16] = vv;
        }
      }
    }
  }
}

// ---- layer 4: out = sum_i W4[i]*src[i] + src[0], 4 columns/thread (b128) ---
__global__ __launch_bounds__(256) void genie_layer4(
    const float* __restrict__ src, const float* __restrict__ W4,
    float* __restrict__ out) {
  const long c = ((long)blockIdx.x * 256 + threadIdx.x) * 4;
  float4 acc = *(const float4*)(src + c);  // residual (channel 0)
#pragma unroll
  for (int i = 0; i < NPATH; ++i) {
    const float w = W4[i];
    const float4 s = *(const float4*)(src + (long)i * NC + c);
    acc.x += w * s.x; acc.y += w * s.y; acc.z += w * s.z; acc.w += w * s.w;
  }
  *(float4*)(out + c) = acc;
}

extern "C" void kernel_launch(void* const* d_in, const int* in_sizes, int n_in,
                              void* d_out, int out_size, void* d_ws, size_t ws_size,
                              hipStream_t stream) {
  (void)in_sizes; (void)n_in; (void)out_size; (void)ws_size;
  const float* v  = (const float*)d_in[0];
  const float* U  = (const float*)d_in[1];
  const float* W0 = (const float*)d_in[2];
  const float* W1 = (const float*)d_in[3];
  const float* W2 = (const float*)d_in[4];
  const float* W3 = (const float*)d_in[5];
  const float* W4 = (const float*)d_in[6];
  float* out = (float*)d_out;

  float* bufA = (float*)d_ws;
  float* bufB = bufA + NPATH * NC;
  float* mats = bufB + NPATH * NC;
  float* wpad = mats + 34L * NSITE * 9;

  // Build path table (mirrors build_paths) + (mu, level, sign) -> path index.
  const int Ldim[4] = {32, 16, 16, 16};
  PathTab pt; pt.mu[0] = 0; pt.n[0] = 0;
  int pidx[4][5][2];
  int nlev[4];
  int idx = 1;
  for (int mu = 0; mu < 4; ++mu) {
    int n = 1, lev = 0;
    while (n <= Ldim[mu] / 2) {
      pidx[mu][lev][0] = idx; pt.mu[idx] = (short)mu; pt.n[idx] = (short)n;  ++idx;
      pidx[mu][lev][1] = idx; pt.mu[idx] = (short)mu; pt.n[idx] = (short)-n; ++idx;
      n *= 2; ++lev;
    }
    nlev[mu] = lev;
  }

  const dim3 b256(256);
  const int sblocks = (int)(NSITE / 256);  // 512

  genie_padw<<<dim3((3 * 48 * 36 + 255) / 256), b256, 0, stream>>>(W1, W2, W3, wpad);

  // Length-1 transports, then power-of-2 doubling (stream order = dependency).
  for (int mu = 0; mu < 4; ++mu)
    genie_matinit<<<dim3(sblocks), b256, 0, stream>>>(
        U, mats + (long)(pidx[mu][0][0] - 1) * NSITE * 9,
           mats + (long)(pidx[mu][0][1] - 1) * NSITE * 9, mu);
  for (int mu = 0; mu < 4; ++mu) {
    int n = 1;
    for (int lev = 0; lev + 1 < nlev[mu]; ++lev, n *= 2) {
      genie_matdouble<<<dim3(sblocks), b256, 0, stream>>>(
          mats + (long)(pidx[mu][lev][0] - 1) * NSITE * 9,
          mats + (long)(pidx[mu][lev + 1][0] - 1) * NSITE * 9, mu, +n);
      genie_matdouble<<<dim3(sblocks), b256, 0, stream>>>(
          mats + (long)(pidx[mu][lev][1] - 1) * NSITE * 9,
          mats + (long)(pidx[mu][lev + 1][1] - 1) * NSITE * 9, mu, -n);
    }
  }

  // Layer 0 (1->35 broadcast, b128) then PT.
  genie_layer0<<<dim3((int)(NC / 4 / 256)), b256, 0, stream>>>(v, W0, bufB);
  genie_transport<<<dim3(sblocks, NPATH), b256, 0, stream>>>(bufB, bufA, mats, pt);

  // Layers 1..3: WMMA dense + PT. 3072 blocks x 8 waves x 64 columns = NC.
  for (int l = 0; l < 3; ++l) {
    genie_dense_wmma<<<dim3((int)(NC / 512)), b256, 0, stream>>>(
        bufA, wpad + l * 1728, bufB);
    genie_transport<<<dim3(sblocks, NPATH), b256, 0, stream>>>(bufB, bufA, mats, pt);
  }

  // Layer 4: 35 -> 1 reduction straight into d_out (b128).
  genie_layer4<<<dim3((int)(NC / 4 / 256)), b256, 0, stream>>>(bufA, W4, out);
}